// DrawImageLayer_57956288692652
// MI455X (gfx1250) — compile-verified
//
#include <hip/hip_runtime.h>

typedef __attribute__((ext_vector_type(2))) float v2f;
typedef __attribute__((ext_vector_type(8))) float v8f;

#define CANVAS 28
#define TSTEPS 64
#define WAVES_PER_BATCH 4
#define BATCH_PER_BLOCK 2   // 256 threads = 8 waves = 2 batches
#define PIX (CANVAS * CANVAS)

__device__ __forceinline__ v8f vmax8(v8f a, v8f b) {
    v8f r;
#pragma unroll
    for (int i = 0; i < 8; ++i) r[i] = fmaxf(a[i], b[i]);
    return r;
}

__global__ __launch_bounds__(256) void draw_strokes_wmma(
    const float* __restrict__ x, float* __restrict__ out, int B)
{
    __shared__ float smax[BATCH_PER_BLOCK * PIX];

    const int tid   = threadIdx.x;
    const int lane  = tid & 31;
    const int wave  = tid >> 5;               // 0..7
    const int bLoc  = wave >> 2;              // 0..1  batch within block
    const int quart = wave & 3;               // 0..3  stroke quarter
    int b = blockIdx.x * BATCH_PER_BLOCK + bLoc;
    const bool bValid = (b < B);
    if (!bValid) b = 0;                        // keep loads in range; EXEC stays all-ones

    // zero-init shared max buffer (strokes are >= 0, matches image0 = 0)
#pragma unroll
    for (int i = tid; i < BATCH_PER_BLOCK * PIX; i += 256) smax[i] = 0.0f;
    __syncthreads();

    const int   l     = lane & 15;
    const bool  hi    = lane >= 16;
    const float g     = (CANVAS * 0.5f) * (CANVAS * 0.5f);     // (SIZE/2)^2 = 196
    const float rngLo = (float)l        * (1.0f / CANVAS) - 0.5f;
    const float rngHi = (float)(l + 16) * (1.0f / CANVAS) - 0.5f;
    const bool  hiOk  = (l + 16) < CANVAS;                     // rows/cols 16..27 valid

    v8f acc00 = {}; v8f acc01 = {}; v8f acc10 = {}; v8f acc11 = {};
    const v8f zero = {};

    const float* xb = x + (size_t)b * (TSTEPS * 3) + quart * (TSTEPS / WAVES_PER_BATCH) * 3;

#pragma unroll 2
    for (int t = 0; t < TSTEPS / WAVES_PER_BATCH; ++t) {
        const float sx = xb[t * 3 + 0];
        const float sy = xb[t * 3 + 1];
        const float si = xb[t * 3 + 2];

        const float dyLo = rngLo - sy, dyHi = rngHi - sy;
        const float dxLo = rngLo - sx, dxHi = rngHi - sx;
        // fold intensity into the A (py) operand: (si*py)*px matches reference order
        const float pyLo = __expf(-(dyLo * dyLo) * g) * si;
        const float pyHi = hiOk ? __expf(-(dyHi * dyHi) * g) * si : 0.0f;
        const float pxLo = __expf(-(dxLo * dxLo) * g);
        const float pxHi = hiOk ? __expf(-(dxHi * dxHi) * g) : 0.0f;

        // A 16x4: (M=m,K=0) lives in lanes 0-15, VGPR0. Zero everything else so
        // only the K=0 rank-1 term contributes. Same for B 4x16 row K=0.
        v2f aLo = { hi ? 0.0f : pyLo, 0.0f };
        v2f aHi = { hi ? 0.0f : pyHi, 0.0f };
        v2f bLo = { hi ? 0.0f : pxLo, 0.0f };
        v2f bHi = { hi ? 0.0f : pxHi, 0.0f };

        v8f d;
        d = __builtin_amdgcn_wmma_f32_16x16x4_f32(false, aLo, false, bLo, (short)0, zero, false, false);
        acc00 = vmax8(acc00, d);
        d = __builtin_amdgcn_wmma_f32_16x16x4_f32(false, aLo, false, bHi, (short)0, zero, false, false);
        acc01 = vmax8(acc01, d);
        d = __builtin_amdgcn_wmma_f32_16x16x4_f32(false, aHi, false, bLo, (short)0, zero, false, false);
        acc10 = vmax8(acc10, d);
        d = __builtin_amdgcn_wmma_f32_16x16x4_f32(false, aHi, false, bHi, (short)0, zero, false, false);
        acc11 = vmax8(acc11, d);
    }

    // Merge per-wave partial maxima with LDS float-max atomics (ds_max_num_f32).
    // C/D layout: VGPR r, lanes 0-15 -> (M=r, N=l); lanes 16-31 -> (M=r+8, N=l-16).
    float* sb = smax + bLoc * PIX;
    const int rbase = hi ? 8 : 0;
#pragma unroll
    for (int r = 0; r < 8; ++r) {
        const int row0 = r + rbase;       // < 16
        const int row1 = row0 + 16;
        const int col0 = l;               // < 16
        const int col1 = l + 16;
        __hip_atomic_fetch_max(&sb[row0 * CANVAS + col0], acc00[r],
                               __ATOMIC_RELAXED, __HIP_MEMORY_SCOPE_WORKGROUP);
        if (col1 < CANVAS)
            __hip_atomic_fetch_max(&sb[row0 * CANVAS + col1], acc01[r],
                                   __ATOMIC_RELAXED, __HIP_MEMORY_SCOPE_WORKGROUP);
        if (row1 < CANVAS)
            __hip_atomic_fetch_max(&sb[row1 * CANVAS + col0], acc10[r],
                                   __ATOMIC_RELAXED, __HIP_MEMORY_SCOPE_WORKGROUP);
        if (row1 < CANVAS && col1 < CANVAS)
            __hip_atomic_fetch_max(&sb[row1 * CANVAS + col1], acc11[r],
                                   __ATOMIC_RELAXED, __HIP_MEMORY_SCOPE_WORKGROUP);
    }
    __syncthreads();

    // Cooperative, fully coalesced store of both canvases with the final clamp.
#pragma unroll
    for (int i = tid; i < BATCH_PER_BLOCK * PIX; i += 256) {
        const int bl  = i / PIX;
        const int idx = i - bl * PIX;
        const int bb  = blockIdx.x * BATCH_PER_BLOCK + bl;
        if (bb < B)
            out[(size_t)bb * PIX + idx] = fminf(smax[i], 1.0f);
    }
}

extern "C" void kernel_launch(void* const* d_in, const int* in_sizes, int n_in,
                              void* d_out, int out_size, void* d_ws, size_t ws_size,
                              hipStream_t stream) {
    const float* x  = (const float*)d_in[0];
    float* out      = (float*)d_out;
    const int B     = in_sizes[0] / (TSTEPS * 3);   // 1024
    dim3 block(256);
    dim3 grid((B + BATCH_PER_BLOCK - 1) / BATCH_PER_BLOCK);  // 512 blocks
    draw_strokes_wmma<<<grid, block, 0, stream>>>(x, out, B);
}